// AugmentedTripletLoss_1932735283478
// MI455X (gfx1250) — compile-verified
//
#include <hip/hip_runtime.h>
#include <math.h>

#define NN 4096
#define DD 768
#define PP 100
#define MARGIN 1.0f

typedef float v2f __attribute__((ext_vector_type(2)));
typedef float v8f __attribute__((ext_vector_type(8)));
typedef unsigned int v4u __attribute__((ext_vector_type(4)));
typedef int v4i __attribute__((ext_vector_type(4)));
typedef int v8i __attribute__((ext_vector_type(8)));

// workspace layout (float units)
#define OFF_SQ    0
#define OFF_CN    (OFF_SQ + NN)            // normalized centers: P*D
#define OFF_CSQ   (OFF_CN + PP*DD)
#define OFF_AP    (OFF_CSQ + 128)          // uint bits, per-row hardest positive
#define OFF_AN    (OFF_AP + NN)            // uint bits, per-row hardest (class) negative
#define OFF_CMIN  (OFF_AN + NN)            // per-row min distance to centers

#define INF_BITS 0x7F800000u

// ---------------------------------------------------------------- reductions
__device__ __forceinline__ float block_reduce_sum(float v, volatile float* red) {
  #pragma unroll
  for (int off = 16; off > 0; off >>= 1) v += __shfl_xor(v, off, 32);
  const int wave = threadIdx.x >> 5;
  if ((threadIdx.x & 31) == 0) red[wave] = v;
  __syncthreads();
  float t = 0.f;
  if (threadIdx.x == 0) {
    const int nw = blockDim.x >> 5;
    for (int w = 0; w < nw; ++w) t += red[w];
    red[0] = t;
  }
  __syncthreads();
  t = red[0];
  __syncthreads();
  return t;
}

// ---------------------------------------------------------------- init
__global__ void init_kernel(unsigned* __restrict__ row_ap, unsigned* __restrict__ row_an) {
  int i = blockIdx.x * blockDim.x + threadIdx.x;
  if (i < NN) {
    row_ap[i] = 0u;        // distances >= 1e-6, 0 is identity for max
    row_an[i] = INF_BITS;  // +inf is identity for min
  }
}

// ---------------------------------------------------------------- ||x_i||^2
__global__ __launch_bounds__(256) void rowsq_kernel(const float* __restrict__ x,
                                                    float* __restrict__ sq) {
  __shared__ float red[8];
  const int row = blockIdx.x;
  const float* r = x + (size_t)row * DD;
  float s = 0.f;
  for (int d = threadIdx.x; d < DD; d += 256) { float v = r[d]; s += v * v; }
  s = block_reduce_sum(s, red);
  if (threadIdx.x == 0) sq[row] = s;
}

// ---------------------------------------------------------------- normalize centers
__global__ __launch_bounds__(256) void centers_kernel(const float* __restrict__ center,
                                                      float* __restrict__ cn,
                                                      float* __restrict__ csq) {
  __shared__ float red[8];
  const int p = blockIdx.x;
  const float* c = center + (size_t)p * DD;
  float s = 0.f;
  for (int d = threadIdx.x; d < DD; d += 256) { float v = c[d]; s += v * v; }
  s = block_reduce_sum(s, red);
  const float rn = 1.0f / sqrtf(s);
  float s2 = 0.f;
  for (int d = threadIdx.x; d < DD; d += 256) {
    float v = c[d] * rn;
    cn[(size_t)p * DD + d] = v;
    s2 += v * v;
  }
  s2 = block_reduce_sum(s2, red);
  if (threadIdx.x == 0) csq[p] = s2;
}

// ---------------------------------------------------------------- fused Gram + hardest pos/neg
// Triangular grid over 128x128 tiles (bi <= bj): row-pass merges rows of tile bi,
// column-pass merges rows of tile bj (transpose). Diagonal tiles need row-pass only.
// Staging: TENSOR_LOAD_TO_LDS (TDM) double-buffered; D# pads 1 DWORD per 16 so the
// LDS layout is the bank-spread stride-17 the WMMA fragment loads expect.
#define BM 128
#define BN 128
#define KC 16
#define LDT 17
#define NTILE (NN / BM)          // 32
#define NCHUNK (DD / KC)         // 48

__device__ __forceinline__ void tdm_load_tile(const float* gbase, int row0, int k0,
                                              unsigned lds_byte) {
  unsigned long long ga = (unsigned long long)(uintptr_t)gbase +
                          ((unsigned long long)row0 * DD + (unsigned)k0) * 4ull;
  v4u g0;
  g0[0] = 1u;                                        // count=1 valid user D#
  g0[1] = lds_byte;                                  // lds_addr (bytes)
  g0[2] = (unsigned)(ga & 0xFFFFFFFFull);            // global_addr[31:0]
  g0[3] = (unsigned)((ga >> 32) & 0x01FFFFFFull) | 0x80000000u;  // addr[56:32] | type=2
  v8i g1;
  g1[0] = (int)((2u << 16) | (1u << 20) | (3u << 22));  // 4B elems, pad_en, 1DW per 16DW
  g1[1] = (int)(((unsigned)DD & 0xFFFFu) << 16);        // tensor_dim0 lo16 (768)
  g1[2] = (int)((((unsigned)DD >> 16) & 0xFFFFu) | (((unsigned)NN & 0xFFFFu) << 16));
  g1[3] = (int)((((unsigned)NN >> 16) & 0xFFFFu) | (16u << 16));  // tile_dim0 = 16
  g1[4] = 128;                                          // tile_dim1 = 128 rows
  g1[5] = DD;                                           // tensor_dim0_stride lo32
  g1[6] = 0;
  g1[7] = 0;
  v4i z4 = {0, 0, 0, 0};
  v8i z8 = {0, 0, 0, 0, 0, 0, 0, 0};
  __builtin_amdgcn_tensor_load_to_lds(g0, g1, z4, z4, z8, 0);
}

__global__ __launch_bounds__(256) void gram_kernel(const float* __restrict__ x,
                                                   const int* __restrict__ tgt,
                                                   const float* __restrict__ sq,
                                                   unsigned* __restrict__ row_ap,
                                                   unsigned* __restrict__ row_an) {
  __shared__ float As[2][BM * LDT];
  __shared__ float Bs[2][BN * LDT];

  const int tid   = threadIdx.x;
  const int wave  = tid >> 5;
  const int lane  = tid & 31;
  const int lm    = lane & 15;           // M (A frag) / N (B frag & C cols)
  const int khalf = (lane >> 4) << 1;    // upper half-wave holds K+2,K+3
  const int mw    = wave >> 1;           // 0..3
  const int nw    = wave & 1;            // 0..1

  // decode triangular block index -> (bi, bj), bi <= bj
  int b  = blockIdx.x;
  int bi = (int)((2.0f * NTILE + 1.0f -
                  sqrtf((float)((2 * NTILE + 1) * (2 * NTILE + 1) - 8 * b))) * 0.5f);
  if (bi < 0) bi = 0;
  if (bi > NTILE - 1) bi = NTILE - 1;
  while (bi > 0 && (bi * NTILE - (bi * (bi - 1)) / 2) > b) --bi;
  while (((bi + 1) * NTILE - ((bi + 1) * bi) / 2) <= b) ++bi;
  const int bj = bi + (b - (bi * NTILE - (bi * (bi - 1)) / 2));
  const int blockM = bi * BM;
  const int blockN = bj * BN;
  const bool offdiag = (bi != bj);

  v8f acc[2][4];
  #pragma unroll
  for (int a = 0; a < 2; ++a)
    #pragma unroll
    for (int c = 0; c < 4; ++c)
      #pragma unroll
      for (int e = 0; e < 8; ++e) acc[a][c][e] = 0.f;

  // prologue: DMA chunk 0 into buffer 0
  if (wave == 0) {
    tdm_load_tile(x, blockM, 0, (unsigned)(uintptr_t)&As[0][0]);
    tdm_load_tile(x, blockN, 0, (unsigned)(uintptr_t)&Bs[0][0]);
  }

  for (int t = 0; t < NCHUNK; ++t) {
    const int buf = t & 1;
    if (wave == 0) {
      if (t + 1 < NCHUNK) {
        const int nb = (t + 1) & 1;
        tdm_load_tile(x, blockM, (t + 1) * KC, (unsigned)(uintptr_t)&As[nb][0]);
        tdm_load_tile(x, blockN, (t + 1) * KC, (unsigned)(uintptr_t)&Bs[nb][0]);
        __builtin_amdgcn_s_wait_tensorcnt(2);   // chunk t resident (in-order per wave)
      } else {
        __builtin_amdgcn_s_wait_tensorcnt(0);
      }
    }
    __syncthreads();   // buffer `buf` ready for everyone

    #pragma unroll
    for (int kk = 0; kk < KC; kk += 4) {
      v2f a[2], bb[4];
      #pragma unroll
      for (int mf = 0; mf < 2; ++mf) {
        const float* p = &As[buf][(mw * 32 + mf * 16 + lm) * LDT + kk + khalf];
        a[mf][0] = p[0]; a[mf][1] = p[1];
      }
      #pragma unroll
      for (int nf = 0; nf < 4; ++nf) {
        const float* p = &Bs[buf][(nw * 64 + nf * 16 + lm) * LDT + kk + khalf];
        bb[nf][0] = p[0]; bb[nf][1] = p[1];
      }
      #pragma unroll
      for (int mf = 0; mf < 2; ++mf)
        #pragma unroll
        for (int nf = 0; nf < 4; ++nf)
          acc[mf][nf] = __builtin_amdgcn_wmma_f32_16x16x4_f32(
              false, a[mf], false, bb[nf], (short)0, acc[mf][nf], false, false);
    }
    __syncthreads();   // all done reading `buf` before TDM overwrites it at t+1
  }

  // ---- fused epilogue: distance + class mask, row pass + (off-diag) column pass ----
  int   tj[4];
  float sqj[4];
  #pragma unroll
  for (int nf = 0; nf < 4; ++nf) {
    const int j = blockN + nw * 64 + nf * 16 + lm;
    tj[nf]  = tgt[j];
    sqj[nf] = sq[j];
  }
  const float FINF = __uint_as_float(INF_BITS);
  float apc[4], anc[4];   // per-column (transpose) partials over this wave's 32 rows
  #pragma unroll
  for (int nf = 0; nf < 4; ++nf) { apc[nf] = 0.f; anc[nf] = FINF; }

  #pragma unroll
  for (int mf = 0; mf < 2; ++mf) {
    #pragma unroll
    for (int r = 0; r < 8; ++r) {
      const int i    = blockM + mw * 32 + mf * 16 + r + ((lane >> 4) << 3);
      const int ti   = tgt[i];
      const float si = sq[i];
      float ap = 0.f, an = FINF;
      #pragma unroll
      for (int nf = 0; nf < 4; ++nf) {
        const float g  = acc[mf][nf][r];
        const float d2 = si + sqj[nf] - 2.0f * g;
        const float dd = sqrtf(fmaxf(d2, 1e-12f));
        const bool same = (ti == tj[nf]);
        if (same) { ap = fmaxf(ap, dd); apc[nf] = fmaxf(apc[nf], dd); }
        else      { an = fminf(an, dd); anc[nf] = fminf(anc[nf], dd); }
      }
      // row reduce across the 16 lanes holding this row (stays within half-wave)
      #pragma unroll
      for (int off = 1; off < 16; off <<= 1) {
        ap = fmaxf(ap, __shfl_xor(ap, off, 32));
        an = fminf(an, __shfl_xor(an, off, 32));
      }
      if (lm == 0) {
        atomicMax(row_ap + i, __float_as_uint(ap));
        atomicMin(row_an + i, __float_as_uint(an));
      }
    }
  }

  if (offdiag) {   // transpose contribution: columns j reduced over this wave's 32 rows
    #pragma unroll
    for (int nf = 0; nf < 4; ++nf) {
      float a = apc[nf], n = anc[nf];
      a = fmaxf(a, __shfl_xor(a, 16, 32));   // combine M halves
      n = fminf(n, __shfl_xor(n, 16, 32));
      if (lane < 16) {
        const int j = blockN + nw * 64 + nf * 16 + lm;
        atomicMax(row_ap + j, __float_as_uint(a));
        atomicMin(row_an + j, __float_as_uint(n));
      }
    }
  }
}

// ---------------------------------------------------------------- min distance to centers
__global__ __launch_bounds__(128) void d2c_kernel(const float* __restrict__ x,
                                                  const float* __restrict__ cn,
                                                  const float* __restrict__ csq,
                                                  const float* __restrict__ sq,
                                                  float* __restrict__ cmin) {
  __shared__ float xr[DD];
  __shared__ float red[128];
  const int i = blockIdx.x;
  for (int d = threadIdx.x; d < DD; d += 128) xr[d] = x[(size_t)i * DD + d];
  __syncthreads();
  float best = __uint_as_float(INF_BITS);
  const float sqi = sq[i];
  for (int p = threadIdx.x; p < PP; p += 128) {
    const float* c = cn + (size_t)p * DD;
    float dot = 0.f;
    #pragma unroll 4
    for (int d = 0; d < DD; ++d) dot += xr[d] * c[d];
    const float d2 = sqi + csq[p] - 2.0f * dot;
    const float dd = fmaxf(sqrtf(fmaxf(d2, 0.0f)), 1e-12f);
    best = fminf(best, dd);
  }
  red[threadIdx.x] = best;
  __syncthreads();
  #pragma unroll
  for (int s = 64; s > 0; s >>= 1) {
    if (threadIdx.x < s) red[threadIdx.x] = fminf(red[threadIdx.x], red[threadIdx.x + s]);
    __syncthreads();
  }
  if (threadIdx.x == 0) cmin[i] = red[0];
}

// ---------------------------------------------------------------- finalize
__global__ __launch_bounds__(256) void finalize_kernel(const unsigned* __restrict__ row_ap,
                                                       const unsigned* __restrict__ row_an,
                                                       const float* __restrict__ cmin,
                                                       float* __restrict__ out) {
  __shared__ float red[8];
  float s = 0.f;
  for (int i = threadIdx.x; i < NN; i += 256) {
    const float ap = __uint_as_float(row_ap[i]);
    float an = (row_an[i] == INF_BITS) ? (ap + MARGIN) : __uint_as_float(row_an[i]);
    an = fminf(an, cmin[i]);
    s += fmaxf(MARGIN + ap - an, 0.f);
  }
  s = block_reduce_sum(s, red);
  if (threadIdx.x == 0) out[0] = s * (1.0f / (float)NN);
}

// ---------------------------------------------------------------- launch
extern "C" void kernel_launch(void* const* d_in, const int* in_sizes, int n_in,
                              void* d_out, int out_size, void* d_ws, size_t ws_size,
                              hipStream_t stream) {
  const float* x      = (const float*)d_in[0];
  const int*   tgt    = (const int*)d_in[1];
  const float* center = (const float*)d_in[2];

  float*    ws     = (float*)d_ws;
  float*    sq     = ws + OFF_SQ;
  float*    cn     = ws + OFF_CN;
  float*    csq    = ws + OFF_CSQ;
  unsigned* row_ap = (unsigned*)(ws + OFF_AP);
  unsigned* row_an = (unsigned*)(ws + OFF_AN);
  float*    cmin   = ws + OFF_CMIN;
  float*    out    = (float*)d_out;

  init_kernel<<<NN / 256, 256, 0, stream>>>(row_ap, row_an);
  rowsq_kernel<<<NN, 256, 0, stream>>>(x, sq);
  centers_kernel<<<PP, 256, 0, stream>>>(center, cn, csq);
  const int ntri = NTILE * (NTILE + 1) / 2;   // 528 triangular tiles
  gram_kernel<<<ntri, 256, 0, stream>>>(x, tgt, sq, row_ap, row_an);
  d2c_kernel<<<NN, 128, 0, stream>>>(x, cn, csq, sq, cmin);
  finalize_kernel<<<1, 256, 0, stream>>>(row_ap, row_an, cmin, out);
}